// GEMV_31112743092623
// MI455X (gfx1250) — compile-verified
//
#include <hip/hip_runtime.h>

// GEMV-ish: out[16 x 8192] = x[16 x 8192] * w[8192 x 8192]^T, all fp32.
// HBM-bound: 256 MB weight stream @ 23.3 TB/s -> ~11 us floor (AI ~ 8 F/B).
// WMMA f32 16x16x4 per K-step of 4; one wave per 16-row M-tile; K split 16
// ways with f32 atomics; x chunk staged to LDS via async-to-LDS (ASYNCcnt).

typedef __attribute__((ext_vector_type(2))) float v2f;
typedef __attribute__((ext_vector_type(8))) float v8f;
typedef __attribute__((ext_vector_type(4))) int v4i;

#define MDIM 8192
#define NDIM 16
#define KDIM 8192
#define KCHUNK 512
#define KSPLIT (KDIM / KCHUNK)              // 16
#define WAVES_PER_BLOCK 4
#define MTILES (MDIM / 16)                  // 512
#define MGROUPS (MTILES / WAVES_PER_BLOCK)  // 128
#define LDS_STRIDE (KCHUNK + 4)             // 516 floats: banks 4r+c, conflict-free b64 reads

// ---- CDNA5 async memory->LDS copy (ASYNCcnt path), builtin with asm fallback ----
__device__ __forceinline__ void async_copy_b128(const float* gsrc, float* ldst) {
#if __has_builtin(__builtin_amdgcn_global_load_async_to_lds_b128)
  typedef __attribute__((address_space(1))) v4i gv4i;   // global (AS1) 16B vector
  typedef __attribute__((address_space(3))) v4i lv4i;   // LDS (AS3) 16B vector
  __builtin_amdgcn_global_load_async_to_lds_b128(
      (gv4i*)gsrc, (lv4i*)ldst, /*offset=*/0, /*cpol=*/0);
#else
  // GV form: LDS byte address in VDST vgpr, 64-bit global address in vaddr.
  // Low 32 bits of a flat shared pointer are the LDS offset (ISA 10.2).
  unsigned lds_addr = (unsigned)(uintptr_t)ldst;
  asm volatile("global_load_async_to_lds_b128 %0, %1, off"
               :: "v"(lds_addr), "v"(gsrc) : "memory");
#endif
}

__device__ __forceinline__ void async_join() {
#if __has_builtin(__builtin_amdgcn_s_wait_asynccnt)
  __builtin_amdgcn_s_wait_asynccnt(0);
#else
  asm volatile("s_wait_asynccnt 0x0" ::: "memory");
#endif
}

__global__ __launch_bounds__(128) void GEMV16_zero(float* __restrict__ out) {
  int i = blockIdx.x * blockDim.x + threadIdx.x;
  if (i < NDIM * MDIM) out[i] = 0.0f;
}

__global__ __launch_bounds__(128) void GEMV16_wmma(const float* __restrict__ x,
                                                   const float* __restrict__ w,
                                                   float* __restrict__ out) {
  __shared__ float sx[16 * LDS_STRIDE];

  const int mgroup = blockIdx.x % MGROUPS;       // consecutive blocks share the x chunk via L2
  const int kchunk = blockIdx.x / MGROUPS;
  const int k0     = kchunk * KCHUNK;
  const int wave   = threadIdx.x >> 5;
  const int lane   = threadIdx.x & 31;
  const int lane16 = lane & 15;
  const int half   = lane >> 4;                  // wave32: lanes 0-15 vs 16-31
  const int m0     = (mgroup * WAVES_PER_BLOCK + wave) * 16;

  // ---- Stage x[0:16, k0:k0+KCHUNK) into LDS: async b128, no VGPR round-trip ----
  for (int i = threadIdx.x; i < 16 * (KCHUNK / 4); i += 128) {
    const int row = i >> 7;                      // / (KCHUNK/4)=128
    const int c4  = (i & 127) << 2;
    async_copy_b128(&x[(size_t)row * KDIM + k0 + c4],
                    &sx[row * LDS_STRIDE + c4]);
  }
  async_join();                                  // this wave's async stores to LDS done
  __syncthreads();                               // make them visible to all 4 waves

  // ---- Per-lane base pointers matching WMMA f32 16x16x4 operand layouts ----
  // A (16x4): lane L -> x[L%16, k + 2*(L/16)], 2 consecutive floats (ds_load_b64)
  // B (4x16): lane L -> w[m0+L%16, k + 2*(L/16)], 2 consecutive floats (global_load_b64)
  const float* axp  = &sx[lane16 * LDS_STRIDE + 2 * half];
  const float* wrow = w + (size_t)(m0 + lane16) * KDIM + k0 + 2 * half;

  v8f acc = {0.f, 0.f, 0.f, 0.f, 0.f, 0.f, 0.f, 0.f};

#pragma unroll 1
  for (int kk = 0; kk < KCHUNK; kk += 32) {
    // Warm L2 ~4 iterations ahead (speculative scope: dropped on bad translation).
    __builtin_prefetch(wrow + kk + 128, 0, 0);
#pragma unroll
    for (int u = 0; u < 8; ++u) {
      const int kc = kk + 4 * u;
      v2f a = *(const v2f*)(axp + kc);           // ds_load_b64 (conflict-free)
      v2f b = *(const v2f*)(wrow + kc);          // global_load_b64, 256B/wave
      acc = __builtin_amdgcn_wmma_f32_16x16x4_f32(
          /*neg_a=*/false, a, /*neg_b=*/false, b,
          /*c_mod=*/(short)0, acc, /*reuse_a=*/false, /*reuse_b=*/false);
    }
  }

  // ---- D layout: VGPR v, lane L -> D[row = v + 8*(L/16), col = L%16] ----
  float* obase = out + (size_t)(8 * half) * MDIM + m0 + lane16;
#pragma unroll
  for (int v = 0; v < 8; ++v) {
    atomicAdd(obase + (size_t)v * MDIM, acc[v]); // global_atomic_add_f32
  }
}

extern "C" void kernel_launch(void* const* d_in, const int* in_sizes, int n_in,
                              void* d_out, int out_size, void* d_ws, size_t ws_size,
                              hipStream_t stream) {
  const float* x = (const float*)d_in[0];       // [16, 8192]
  const float* w = (const float*)d_in[1];       // [8192, 8192]
  float* out = (float*)d_out;                   // [16, 8192]

  GEMV16_zero<<<(NDIM * MDIM + 127) / 128, 128, 0, stream>>>(out);
  GEMV16_wmma<<<MGROUPS * KSPLIT, 128, 0, stream>>>(x, w, out);
}